// Sampler_50946902065701
// MI455X (gfx1250) — compile-verified
//
#include <hip/hip_runtime.h>

// CDNA5 (gfx1250) implementation of the Gaussian scan sampler.
// Phase 1 (mats_kernel, parallel over t):  Prec=inv(P[t-1]) (Gauss-Jordan in
//   LDS), PG = P[t]@G^T, A = PG@Prec, S = P[t]-A@PG^T+jI, L = chol(S) -> ws.
//   All three 128^3 GEMMs run on V_WMMA_F32_16X16X4_F32.
// Phase 2 (b_kernel, parallel over t):     b[t] = (m[t]-A@m[t-1]) + L@eps^T.
// Phase 3 (scan_kernel, parallel over N):  x_t = A_t x_{t-1} + b_t; A_{t+1}
//   is prefetched into a double-buffered LDS slot with
//   GLOBAL_LOAD_ASYNC_TO_LDS_B128 (ASYNCcnt) to hide the 64KB/step fetch
//   behind the WMMA matmul of the current step.

#define DD 128
#define TT 1024
#define NN 128
#define JIT 1e-4f

typedef __attribute__((ext_vector_type(2))) float v2f;
typedef __attribute__((ext_vector_type(8))) float v8f;

// ---- WMMA f32 16x16x4 lane-layout helpers (ISA 7.12.2) -----------------
// A 16x4: lanes 0-15 M=0..15 {K0,K1}; lanes 16-31 M=0..15 {K2,K3}
static __device__ __forceinline__ v2f load_a_tile(const float* M, int ld,
                                                  int r0, int k0, int lane) {
  int m = lane & 15, kh = lane >> 4;
  const float* p = M + (size_t)(r0 + m) * ld + k0 + 2 * kh;
  v2f a; a.x = p[0]; a.y = p[1];
  return a;
}
// B 4x16 row-major source: B(k,c) = M[k*ld + c]
static __device__ __forceinline__ v2f load_b_tile_n(const float* M, int ld,
                                                    int k0, int c0, int lane) {
  int c = lane & 15, kh = lane >> 4;
  v2f b;
  b.x = M[(size_t)(k0 + 2 * kh + 0) * ld + c0 + c];
  b.y = M[(size_t)(k0 + 2 * kh + 1) * ld + c0 + c];
  return b;
}
// B 4x16 transposed source: B(k,c) = M[c*ld + k]  (for X @ M^T)
static __device__ __forceinline__ v2f load_b_tile_t(const float* M, int ld,
                                                    int k0, int c0, int lane) {
  int c = lane & 15, kh = lane >> 4;
  const float* p = M + (size_t)(c0 + c) * ld + k0 + 2 * kh;
  v2f b; b.x = p[0]; b.y = p[1];
  return b;
}

static __device__ __forceinline__ v8f wmma_f32(v2f a, v2f b, v8f c) {
  return __builtin_amdgcn_wmma_f32_16x16x4_f32(false, a, false, b,
                                               (short)0, c, false, false);
}

// ---- async global -> LDS staging (GLOBAL_LOAD_ASYNC_TO_LDS_B128) -------
// LDS operand = low 32 bits of the generic pointer (LDS aperture puts the
// byte offset in addr[31:0], ISA 10.2).  Tracked by ASYNCcnt.
static __device__ __forceinline__ void async_copy_b128(float* lds_dst,
                                                       const float* gsrc) {
  uint32_t lds_off = (uint32_t)(uintptr_t)lds_dst;
  asm volatile("global_load_async_to_lds_b128 %0, %1, off"
               :: "v"(lds_off), "v"(gsrc)
               : "memory");
}
static __device__ __forceinline__ void wait_asynccnt0() {
  asm volatile("s_wait_asynccnt 0x0" ::: "memory");
}
// stage one 128x128 f32 matrix: 256 threads x 16 issues x 16B, coalesced
static __device__ __forceinline__ void prefetch_mat(const float* __restrict__ src,
                                                    float* dst, int tid) {
#pragma unroll
  for (int j = 0; j < 16; ++j) {
    int e = (j * 256 + tid) * 4;
    async_copy_b128(dst + e, src + e);
  }
}

// ---- 128x128x128 fp32 GEMM on one 256-thread (8-wave) block ------------
// mode 0: C = A*B     mode 1: C = E - A*B + JIT*I
static __device__ void gemm128(const float* A, const float* B, float* C,
                               int transB, int mode, const float* E,
                               int wave, int lane) {
  for (int tile = wave; tile < 64; tile += 8) {
    int r0 = (tile >> 3) << 4;
    int c0 = (tile & 7) << 4;
    v8f acc = {0.f, 0.f, 0.f, 0.f, 0.f, 0.f, 0.f, 0.f};
#pragma unroll 8
    for (int k0 = 0; k0 < DD; k0 += 4) {
      v2f a = load_a_tile(A, DD, r0, k0, lane);
      v2f b = transB ? load_b_tile_t(B, DD, k0, c0, lane)
                     : load_b_tile_n(B, DD, k0, c0, lane);
      acc = wmma_f32(a, b, acc);
    }
    int col = lane & 15, kh = lane >> 4;
#pragma unroll
    for (int v = 0; v < 8; ++v) {
      int r = r0 + v + 8 * kh, c = c0 + col;
      float val = acc[v];
      if (mode == 1) val = E[r * DD + c] - val + ((r == c) ? JIT : 0.f);
      C[r * DD + c] = val;
    }
  }
}

// ---- in-place Cholesky (lower) of SPD 128x128 in LDS, zero upper -------
static __device__ void cholesky128(float* S, int tid) {
  for (int k = 0; k < DD; ++k) {
    if (tid == 0) S[k * DD + k] = sqrtf(S[k * DD + k]);
    __syncthreads();
    float dk = S[k * DD + k];
    for (int i = k + 1 + tid; i < DD; i += 256) S[i * DD + k] /= dk;
    __syncthreads();
    for (int idx = tid; idx < DD * DD; idx += 256) {
      int i = idx >> 7, j = idx & 127;
      if (i > k && j > k && j <= i) S[i * DD + j] -= S[i * DD + k] * S[j * DD + k];
    }
    __syncthreads();
  }
  for (int idx = tid; idx < DD * DD; idx += 256) {
    int i = idx >> 7, j = idx & 127;
    if (j > i) S[idx] = 0.f;
  }
}

// ======================= Phase 1: per-t matrices ========================
__global__ void __launch_bounds__(256)
mats_kernel(const float* __restrict__ P, const float* __restrict__ G,
            float* __restrict__ Aws, float* __restrict__ Lws) {
  extern __shared__ float lds[];
  float* buf0 = lds;                 // P_prev -> junk -> P_t
  float* buf1 = lds + 16384;         // I -> Prec -> S -> L
  float* buf2 = lds + 2 * 16384;     // fcol scratch -> PG
  float* buf3 = lds + 3 * 16384;     // G staging -> A
  int t = blockIdx.x;
  int tid = threadIdx.x, wave = tid >> 5, lane = tid & 31;

  if (t == 0) {                      // x0 path: only L0 = chol(P0), A0 = 0
    for (int i = tid; i < DD * DD; i += 256) buf1[i] = P[i];
    __syncthreads();
    cholesky128(buf1, tid);
    __syncthreads();
    for (int i = tid; i < DD * DD; i += 256) { Lws[i] = buf1[i]; Aws[i] = 0.f; }
    return;
  }
  const float* Pp = P + (size_t)(t - 1) * DD * DD;
  const float* Pt = P + (size_t)t * DD * DD;
  const float* Gp = G + (size_t)(t - 1) * DD * DD;

  for (int i = tid; i < DD * DD; i += 256) {
    buf0[i] = Pp[i];
    int r = i >> 7, c = i & 127;
    buf1[i] = (r == c) ? 1.f : 0.f;
  }
  // stage G[t-1] into LDS asynchronously while Gauss-Jordan runs
  prefetch_mat(Gp, buf3, tid);
  __syncthreads();

  // Gauss-Jordan (no pivoting; P is SPD) -> buf1 = inv(P_prev)
  for (int k = 0; k < DD; ++k) {
    float ip = 1.0f / buf0[k * DD + k];
    __syncthreads();
    for (int j = tid; j < DD; j += 256) {
      buf0[k * DD + j] *= ip;
      buf1[k * DD + j] *= ip;
    }
    __syncthreads();
    for (int i = tid; i < DD; i += 256) buf2[i] = buf0[i * DD + k];
    __syncthreads();
    for (int idx = tid; idx < DD * DD; idx += 256) {
      int i = idx >> 7, j = idx & 127;
      if (i != k) {
        float f = buf2[i];
        buf0[i * DD + j] -= f * buf0[k * DD + j];
        buf1[i * DD + j] -= f * buf1[k * DD + j];
      }
    }
    __syncthreads();
  }

  for (int i = tid; i < DD * DD; i += 256) buf0[i] = Pt[i];
  wait_asynccnt0();                  // G staging complete (per wave)
  __syncthreads();
  gemm128(buf0, buf3, buf2, /*transB=*/1, 0, nullptr, wave, lane); // PG = Pt @ G^T
  __syncthreads();
  gemm128(buf2, buf1, buf3, /*transB=*/0, 0, nullptr, wave, lane); // A  = PG @ Prec
  __syncthreads();
  gemm128(buf3, buf2, buf1, /*transB=*/1, 1, buf0,    wave, lane); // S  = Pt - A@PG^T + jI
  __syncthreads();
  cholesky128(buf1, tid);                                          // L  = chol(S)
  __syncthreads();

  float* At = Aws + (size_t)t * DD * DD;
  float* Lt = Lws + (size_t)t * DD * DD;
  for (int i = tid; i < DD * DD; i += 256) { At[i] = buf3[i]; Lt[i] = buf1[i]; }
}

// ================ Phase 2: b[t] = (m_t - A m_{t-1}) + L eps^T ===========
__global__ void __launch_bounds__(256)
b_kernel(const float* __restrict__ m, const float* __restrict__ eps,
         const float* __restrict__ Aws, const float* __restrict__ Lws,
         float* __restrict__ Bws) {
  extern __shared__ float lds[];
  float* Lbuf = lds;                 // 128x128
  float* Ebuf = lds + 16384;         // eps[t] (N x D) row-major
  float* cvec = lds + 2 * 16384;     // 128
  int t = blockIdx.x, tid = threadIdx.x, wave = tid >> 5, lane = tid & 31;

  const float* Lt = Lws + (size_t)t * DD * DD;
  const float* Et = eps + (size_t)t * NN * DD;
  for (int i = tid; i < DD * DD; i += 256) { Lbuf[i] = Lt[i]; Ebuf[i] = Et[i]; }
  if (tid < DD) {
    float c = m[(size_t)t * DD + tid];
    if (t > 0) {
      const float* At = Aws + (size_t)t * DD * DD;
      const float* mp = m + (size_t)(t - 1) * DD;
      float s = 0.f;
      for (int k = 0; k < DD; ++k) s += At[tid * DD + k] * mp[k];
      c -= s;
    }
    cvec[tid] = c;
  }
  __syncthreads();

  // B_t(d,n) = cvec[d] + sum_k L(d,k) * eps(n,k)   (B-operand transposed)
  float* Bt = Bws + (size_t)t * DD * NN;
  for (int tile = wave; tile < 64; tile += 8) {
    int r0 = (tile >> 3) << 4, c0 = (tile & 7) << 4;
    v8f acc = {0.f, 0.f, 0.f, 0.f, 0.f, 0.f, 0.f, 0.f};
#pragma unroll 8
    for (int k0 = 0; k0 < DD; k0 += 4) {
      v2f a = load_a_tile(Lbuf, DD, r0, k0, lane);
      v2f b = load_b_tile_t(Ebuf, DD, k0, c0, lane);
      acc = wmma_f32(a, b, acc);
    }
    int col = lane & 15, kh = lane >> 4;
#pragma unroll
    for (int v = 0; v < 8; ++v) {
      int r = r0 + v + 8 * kh;
      Bt[r * NN + c0 + col] = acc[v] + cvec[r];
    }
  }
}

// ============ Phase 3: sequential scan, parallel over samples ===========
// Block b owns sample columns [16b, 16b+16); x_t = A_t x_{t-1} + b_t.
// A_{t+1} is async-prefetched into the alternate LDS buffer during step t.
__global__ void __launch_bounds__(256)
scan_kernel(const float* __restrict__ Aws, const float* __restrict__ Bws,
            float* __restrict__ out) {
  extern __shared__ float lds[];
  float* Abuf0 = lds;                // 128x128 (steps with t even.. buf index t&1)
  float* Abuf1 = lds + 16384;        // 128x128
  float* xbuf  = lds + 2 * 16384;    // 128 x 16  (d, local col)
  int tid = threadIdx.x, wave = tid >> 5, lane = tid & 31;
  int c0 = blockIdx.x * 16;          // global sample-column base

  // kick off A[1] into buf1 while we initialize x
  prefetch_mat(Aws + (size_t)1 * DD * DD, Abuf1, tid);

  // x_0 = b_0  (b_kernel folded chol(P0)@eps0 + m0 into Bws[0])
  for (int i = tid; i < DD * 16; i += 256) {
    int d = i >> 4, j = i & 15;
    float v = Bws[(size_t)d * NN + c0 + j];
    xbuf[d * 16 + j] = v;
    out[(size_t)(c0 + j) * (TT * DD) + d] = v;   // out is (N, T, D)
  }
  wait_asynccnt0();
  __syncthreads();

  for (int t = 1; t < TT; ++t) {
    float* Acur = (t & 1) ? Abuf1 : Abuf0;
    float* Anxt = (t & 1) ? Abuf0 : Abuf1;
    // prefetch A[t+1] into the buffer last read at step t-1 (fenced by the
    // barrier at the end of that step)
    if (t + 1 < TT) prefetch_mat(Aws + (size_t)(t + 1) * DD * DD, Anxt, tid);

    const float* Bt = Bws + (size_t)t * DD * NN;
    int r0 = wave << 4;                 // each wave: one 16x16 output tile
    v8f acc = {0.f, 0.f, 0.f, 0.f, 0.f, 0.f, 0.f, 0.f};
#pragma unroll 8
    for (int k0 = 0; k0 < DD; k0 += 4) {
      v2f a = load_a_tile(Acur, DD, r0, k0, lane);
      v2f b = load_b_tile_n(xbuf, 16, k0, 0, lane);
      acc = wmma_f32(a, b, acc);
    }
    int col = lane & 15, kh = lane >> 4;
    float vals[8];
#pragma unroll
    for (int v = 0; v < 8; ++v) {
      int r = r0 + v + 8 * kh;
      vals[v] = acc[v] + Bt[(size_t)r * NN + c0 + col];
    }
    __syncthreads();                    // everyone done reading xbuf/Acur
#pragma unroll
    for (int v = 0; v < 8; ++v) {
      int r = r0 + v + 8 * kh;
      xbuf[r * 16 + col] = vals[v];
      out[(size_t)(c0 + col) * (TT * DD) + (size_t)t * DD + r] = vals[v];
    }
    wait_asynccnt0();                   // own wave's A[t+1] slices landed
    __syncthreads();                    // -> all waves' slices landed
  }
}

// ============================ launcher ==================================
extern "C" void kernel_launch(void* const* d_in, const int* in_sizes, int n_in,
                              void* d_out, int out_size, void* d_ws, size_t ws_size,
                              hipStream_t stream) {
  (void)in_sizes; (void)n_in; (void)out_size; (void)ws_size;
  const float* m   = (const float*)d_in[0];   // (T, D, 1)
  const float* P   = (const float*)d_in[1];   // (T, D, D)
  const float* G   = (const float*)d_in[2];   // (T-1, D, D)
  const float* eps = (const float*)d_in[3];   // (T, N, D, 1)
  float* out = (float*)d_out;                 // (N, T, D)

  float* Aws = (float*)d_ws;                          // T * D*D
  float* Lws = Aws + (size_t)TT * DD * DD;            // T * D*D
  float* Bws = Lws + (size_t)TT * DD * DD;            // T * D*N

  mats_kernel<<<TT, 256, 4 * DD * DD * sizeof(float), stream>>>(P, G, Aws, Lws);
  b_kernel<<<TT, 256, (2 * DD * DD + DD) * sizeof(float), stream>>>(m, eps, Aws, Lws, Bws);
  scan_kernel<<<NN / 16, 256, (2 * DD * DD + DD * 16) * sizeof(float), stream>>>(Aws, Bws, out);
}